// S4Model2D_78709570666705
// MI455X (gfx1250) — compile-verified
//
#include <hip/hip_runtime.h>
#include <math.h>

// ---------------------------------------------------------------------------
// S4ND-2D model for MI455X (gfx1250).  Separable DPLR bidirectional conv
// rewritten as per-channel dense 128x128 matrices on each axis; everything
// runs as batched f32 GEMMs on V_WMMA_F32_16X16X4_F32 (wave32 WMMA).
// Per-wave 32x64 output block (2x4 WMMA tiles); B staged in LDS k-pair
// interleaved so each B fragment is one conflict-free ds_load_b64.
// Workspace use: ~145 MB.
// ---------------------------------------------------------------------------

typedef __attribute__((ext_vector_type(2))) float v2f;
typedef __attribute__((ext_vector_type(4))) float v4f;
typedef __attribute__((ext_vector_type(8))) float v8f;

#define Bn   8
#define Dm   128
#define HW   16384L   // 128*128
#define Nm   32       // modes
#define Ff   65       // L/2+1

// ---------------- small helpers ----------------
struct c2 { float re, im; };
__device__ inline c2 cmul(c2 a, c2 b) { return { a.re*b.re - a.im*b.im, a.re*b.im + a.im*b.re }; }
__device__ inline c2 cdivc(c2 a, c2 b) {
  float inv = 1.0f / (b.re*b.re + b.im*b.im);
  return { (a.re*b.re + a.im*b.im)*inv, (a.im*b.re - a.re*b.im)*inv };
}
__device__ inline float gelu_tanh(float x) {
  float x3 = x*x*x;
  return 0.5f * x * (1.0f + tanhf(0.7978845608028654f * (x + 0.044715f * x3)));
}

// ---------------- DPLR bidirectional kernel tables ----------------
// grid: (cin, 2 axes), 256 threads. Output khat[axis][c][256]:
//   khat[0..127] = K_fwd, khat[128+s] = K_bwd[127-s]
__global__ void kern_build(const float* __restrict__ log_dt, const float* __restrict__ logA,
                           const float* __restrict__ Aim,    const float* __restrict__ Pre,
                           const float* __restrict__ Pim,    const float* __restrict__ Bre,
                           const float* __restrict__ Bim,    const float* __restrict__ Cre,
                           const float* __restrict__ Cim,    int cin,
                           float* khat_h, float* khat_w) {
  int c = blockIdx.x, axis = blockIdx.y, tid = threadIdx.x;
  float* khat = axis ? khat_w : khat_h;
  __shared__ float skr[2][Ff], ski[2][Ff];

  if (tid < 2 * Ff) {
    int dir = tid / Ff, f = tid - dir * Ff;
    float dt = expf(log_dt[axis * cin + c]);
    // z = exp(-2*pi*i*f/L)
    float a = (float)f / 64.0f;            // f/64 = 2f/L
    float zre = cospif(a), zim = -sinpif(a);
    c2 onePz = { 1.0f + zre, zim };
    c2 oneMz = { 1.0f - zre, -zim };
    c2 q = cdivc(oneMz, onePz);
    float s2dt = 2.0f / dt;
    float gre = s2dt * q.re, gim = s2dt * q.im;

    c2 s00 = {0,0}, s01 = {0,0}, s10 = {0,0}, s11 = {0,0};
    for (int n = 0; n < Nm; ++n) {
      int idx = (axis * cin + c) * Nm + n;
      float wre = -expf(logA[idx]);
      float wim = Aim[idx];
      float dre = gre - wre, dim_ = gim - wim;
      float inv = 1.0f / (dre*dre + dim_*dim_);
      c2 invden = { dre * inv, -dim_ * inv };
      c2 P  = { Pre[idx], Pim[idx] };
      c2 Pc = { P.re, -P.im };
      c2 Bv = { Bre[idx], Bim[idx] };
      int cidx = ((axis * 2 + dir) * cin + c) * Nm + n;
      c2 Cv = { Cre[cidx], Cim[cidx] };
      c2 t;
      t = cmul(cmul(Cv, Bv), invden); s00.re += t.re; s00.im += t.im;
      t = cmul(cmul(Cv, P ), invden); s01.re += t.re; s01.im += t.im;
      t = cmul(cmul(Pc, Bv), invden); s10.re += t.re; s10.im += t.im;
      t = cmul(cmul(Pc, P ), invden); s11.re += t.re; s11.im += t.im;
    }
    c2 denom = { 1.0f + s11.re, s11.im };
    c2 corr = cdivc(cmul(s01, s10), denom);
    c2 kk = { s00.re - corr.re, s00.im - corr.im };
    c2 two = { 2.0f, 0.0f };
    c2 kf = cmul(kk, cdivc(two, onePz));
    skr[dir][f] = kf.re; ski[dir][f] = kf.im;
  }
  __syncthreads();

  // direct irfft (L=128): K[t] = (X0 + (-1)^t X64 + 2*sum_{f=1..63}(Xre cos - Xim sin)) / L
  {
    int dir = tid >> 7, t = tid & 127;
    float acc = skr[dir][0] + ((t & 1) ? -skr[dir][64] : skr[dir][64]);
    for (int f = 1; f < 64; ++f) {
      float ang = (float)(f * t) / 64.0f;  // (2*pi*f*t/L)/pi
      float co = cospif(ang), si = sinpif(ang);
      acc += 2.0f * (skr[dir][f] * co - ski[dir][f] * si);
    }
    acc *= (1.0f / 128.0f);
    if (dir == 0) khat[c * 256 + t] = acc;
    else          khat[c * 256 + 128 + (127 - t)] = acc;
  }
}

// Mh[c][i][p] = kh[(i-p)&255];  MwT[c][q][j] = kw[(j-q)&255]
__global__ void mat_build(const float* __restrict__ kh, const float* __restrict__ kw,
                          float* __restrict__ Mh, float* __restrict__ MwT) {
  int c = blockIdx.x;
  for (int idx = threadIdx.x; idx < 16384; idx += 256) {
    int row = idx >> 7, col = idx & 127;
    Mh [c * 16384 + idx] = kh[c * 256 + ((row - col) & 255)];
    MwT[c * 16384 + idx] = kw[c * 256 + ((col - row) & 255)];
  }
}

// ---------------- WMMA GEMM: C(128 x 128cols-per-ny) = A(128xK) * B(Kx...) ----
// 256 threads = 8 wave32; wave w owns a 32x64 block: rows 32*(w>>1), cols 64*(w&1)
// = 2x4 tiles of 16x16.  EPI: 0 plain, 1 gelu(acc + D[zc]*Xres), 2 acc+bias(+Res)
#define SA_S 36    // LDS stride for A chunk (bank-conflict padding)
#define SB_PS 288  // LDS floats per B k-pair row; 288 % 64 == 32 (half-waves disjoint)
template<int EPI>
__global__ __launch_bounds__(256) void gemm128(
    const float* __restrict__ A, const float* Bm, float* Cm_,
    int Cch,
    long sAb, long sAc, long sBb, long sBc, long sCb, long sCc,
    int lda, int ldb, int ldc, int K,
    const float* Dskip, const float* Xres, long sXb, long sXc,
    const float* bias, const float* Res) {
  __shared__ __align__(16) float sA[128 * SA_S];
  __shared__ __align__(16) float sB[16 * SB_PS];  // (k,n) at sB[(k>>1)*288 + 2n + (k&1)]
  int z = blockIdx.x, ny = blockIdx.y;
  int zb = z / Cch, zc = z - zb * Cch;
  long ncol = (long)ny * 128;
  const float* Ap = A   + (long)zb * sAb + (long)zc * sAc;
  const float* Bp = Bm  + (long)zb * sBb + (long)zc * sBc + ncol;
  float*       Cp = Cm_ + (long)zb * sCb + (long)zc * sCc + ncol;

  int tid = threadIdx.x;
  int wave = tid >> 5, lane = tid & 31;
  int lh = lane & 15;
  int khi = (lane >> 4) << 1;     // 0 for lanes 0-15, 2 for lanes 16-31
  int wr = (wave >> 1) << 5;      // row base: 0,32,64,96
  int wc = (wave & 1) << 6;       // col base: 0,64

  v8f acc[2][4];
  for (int i = 0; i < 2; ++i)
    for (int j = 0; j < 4; ++j)
      for (int q = 0; q < 8; ++q) acc[i][j][q] = 0.0f;

  // one k-step: 2 A frags + 4 B frags (single b64 each) -> 8 WMMAs
  auto kstep = [&](int kk) {
    v2f a0 = *(const v2f*)(&sA[(wr + lh)      * SA_S + kk + khi]);
    v2f a1 = *(const v2f*)(&sA[(wr + 16 + lh) * SA_S + kk + khi]);
    int pbase = ((kk + khi) >> 1) * SB_PS;
    v2f bf[4];
    #pragma unroll
    for (int j = 0; j < 4; ++j) {
      int col = wc + (j << 4) + lh;
      bf[j] = *(const v2f*)(&sB[pbase + (col << 1)]);
    }
    #pragma unroll
    for (int j = 0; j < 4; ++j) {
      acc[0][j] = __builtin_amdgcn_wmma_f32_16x16x4_f32(
          false, a0, false, bf[j], (short)0, acc[0][j], false, false);
      acc[1][j] = __builtin_amdgcn_wmma_f32_16x16x4_f32(
          false, a1, false, bf[j], (short)0, acc[1][j], false, false);
    }
  };

  for (int kt = 0; kt < K; kt += 32) {
    bool full = (K - kt) >= 32;
    if (full && ((lda & 3) == 0)) {          // vector stage A: 128 rows x 32 k
      for (int i = tid; i < 1024; i += 256) {
        int r = i >> 3, s = (i & 7) << 2;
        *(v4f*)(&sA[r * SA_S + s]) = *(const v4f*)(Ap + (long)r * lda + kt + s);
      }
    } else {
      for (int i = tid; i < 128 * 32; i += 256) {
        int r = i >> 5, k = i & 31;
        sA[r * SA_S + k] = (kt + k < K) ? Ap[(long)r * lda + kt + k] : 0.0f;
      }
    }
    if (full && ((ldb & 3) == 0)) {          // vector stage B, pair-interleaved
      for (int i = tid; i < 512; i += 256) {
        int p = i >> 5, n4 = (i & 31) << 2;  // k-pair, col group of 4
        v4f r0 = *(const v4f*)(Bp + (long)(kt + 2 * p)     * ldb + n4);
        v4f r1 = *(const v4f*)(Bp + (long)(kt + 2 * p + 1) * ldb + n4);
        v4f w0 = { r0.x, r1.x, r0.y, r1.y };
        v4f w1 = { r0.z, r1.z, r0.w, r1.w };
        int base = p * SB_PS + (n4 << 1);
        *(v4f*)(&sB[base])     = w0;
        *(v4f*)(&sB[base + 4]) = w1;
      }
    } else {
      for (int i = tid; i < 32 * 128; i += 256) {
        int k = i >> 7, n = i & 127;
        sB[(k >> 1) * SB_PS + (n << 1) + (k & 1)] =
            (kt + k < K) ? Bp[(long)(kt + k) * ldb + n] : 0.0f;
      }
    }
    __syncthreads();
    if (full) {
      #pragma unroll
      for (int kk = 0; kk < 32; kk += 4) kstep(kk);
    } else {
      int kmax = ((K - kt) + 3) & ~3;
      for (int kk = 0; kk < kmax; kk += 4) kstep(kk);
    }
    __syncthreads();
  }

  // C/D 16x16 f32 layout: vgpr v -> row = v + (lane<16?0:8), col = lane&15
  int rsub = (lane >> 4) << 3;
  float dcoef = 0.0f;
  const float* Xp = nullptr;
  if (EPI == 1) { dcoef = Dskip[zc]; Xp = Xres + (long)zb * sXb + (long)zc * sXc + ncol; }
  const float* Rp = (EPI == 2 && Res) ? (Res + (long)zb * sCb + (long)zc * sCc + ncol) : nullptr;
  for (int i = 0; i < 2; ++i) {
    for (int j = 0; j < 4; ++j) {
      int col = wc + (j << 4) + lh;
      for (int v = 0; v < 8; ++v) {
        int row = wr + (i << 4) + rsub + v;
        float val = acc[i][j][v];
        if (EPI == 1) val = gelu_tanh(val + dcoef * Xp[(long)row * ldc + col]);
        if (EPI == 2) { val += bias[row]; if (Rp) val += Rp[(long)row * ldc + col]; }
        Cp[(long)row * ldc + col] = val;
      }
    }
  }
}

// ---------------- BatchNorm2d (in-place, biased var) ----------------
__global__ void bn2d(float* X, const float* __restrict__ g, const float* __restrict__ b) {
  int c = blockIdx.x, tid = threadIdx.x;
  double s = 0.0, s2 = 0.0;
  for (int i = tid; i < Bn * 16384; i += 256) {
    int bb = i >> 14, sp = i & 16383;
    float v = X[((long)bb * Dm + c) * HW + sp];
    s += v; s2 += (double)v * v;
  }
  __shared__ double r1[256], r2[256];
  r1[tid] = s; r2[tid] = s2;
  __syncthreads();
  for (int off = 128; off > 0; off >>= 1) {
    if (tid < off) { r1[tid] += r1[tid + off]; r2[tid] += r2[tid + off]; }
    __syncthreads();
  }
  __shared__ float mu_sh, rs_sh;
  if (tid == 0) {
    double mu = r1[0] / 131072.0;
    double var = r2[0] / 131072.0 - mu * mu;
    mu_sh = (float)mu;
    rs_sh = (float)(1.0 / sqrt(var + 1e-5));
  }
  __syncthreads();
  float scale = rs_sh * g[c];
  float shift = b[c] - mu_sh * scale;
  for (int i = tid; i < Bn * 16384; i += 256) {
    int bb = i >> 14, sp = i & 16383;
    long idx = ((long)bb * Dm + c) * HW + sp;
    X[idx] = X[idx] * scale + shift;
  }
}

// ---------------- final spatial mean pool ----------------
__global__ void pool_mean(const float* __restrict__ X, float* __restrict__ out) {
  int bc = blockIdx.x, tid = threadIdx.x;
  float s = 0.0f;
  for (int i = tid; i < 16384; i += 256) s += X[(long)bc * HW + i];
  __shared__ float red[256];
  red[tid] = s;
  __syncthreads();
  for (int off = 128; off > 0; off >>= 1) {
    if (tid < off) red[tid] += red[tid + off];
    __syncthreads();
  }
  if (tid == 0) out[bc] = red[0] * (1.0f / 16384.0f);
}

// ---------------------------------------------------------------------------
extern "C" void kernel_launch(void* const* d_in, const int* in_sizes, int n_in,
                              void* d_out, int out_size, void* d_ws, size_t ws_size,
                              hipStream_t stream) {
  (void)in_sizes; (void)n_in; (void)out_size; (void)ws_size;
  const float* x        = (const float*)d_in[0];
  const float* f_log_dt = (const float*)d_in[1];
  const float* f_logA   = (const float*)d_in[2];
  const float* f_Aim    = (const float*)d_in[3];
  const float* f_Pre    = (const float*)d_in[4];
  const float* f_Pim    = (const float*)d_in[5];
  const float* f_Bre    = (const float*)d_in[6];
  const float* f_Bim    = (const float*)d_in[7];
  const float* f_Cre    = (const float*)d_in[8];
  const float* f_Cim    = (const float*)d_in[9];
  const float* f_D      = (const float*)d_in[10];
  const float* f_W      = (const float*)d_in[11];
  const float* f_b      = (const float*)d_in[12];
  const float* l_log_dt = (const float*)d_in[13];
  const float* l_logA   = (const float*)d_in[14];
  const float* l_Aim    = (const float*)d_in[15];
  const float* l_Pre    = (const float*)d_in[16];
  const float* l_Pim    = (const float*)d_in[17];
  const float* l_Bre    = (const float*)d_in[18];
  const float* l_Bim    = (const float*)d_in[19];
  const float* l_Cre    = (const float*)d_in[20];
  const float* l_Cim    = (const float*)d_in[21];
  const float* l_D      = (const float*)d_in[22];
  const float* l_W      = (const float*)d_in[23];
  const float* l_b      = (const float*)d_in[24];
  const float* bn_g     = (const float*)d_in[25];
  const float* bn_b     = (const float*)d_in[26];

  float* ws   = (float*)d_ws;
  float* Pa   = ws;                         // 8*128*128*128 = 16.78M floats
  float* Pb   = ws + 16777216L;             // same size
  float* Mh   = ws + 33554432L;             // 128*128*128 = 2.10M floats
  float* MwT  = Mh + 2097152L;
  float* khh  = MwT + 2097152L;             // 128*256
  float* khw  = khh + 32768L;

  // ---- first layer: cin = 3 ----
  kern_build<<<dim3(3, 2), 256, 0, stream>>>(f_log_dt, f_logA, f_Aim, f_Pre, f_Pim,
                                             f_Bre, f_Bim, f_Cre, f_Cim, 3, khh, khw);
  mat_build<<<3, 256, 0, stream>>>(khh, khw, Mh, MwT);
  // T(Pb) = X @ MwT
  gemm128<0><<<dim3(24, 1), 256, 0, stream>>>(x, MwT, Pb, 3,
      3*HW, HW, 0, HW, 3*HW, HW, 128, 128, 128, 128,
      nullptr, nullptr, 0, 0, nullptr, nullptr);
  // Pb = gelu(Mh @ Pb + D*X)  (in-place; each block reads only its own tile)
  gemm128<1><<<dim3(24, 1), 256, 0, stream>>>(Mh, Pb, Pb, 3,
      0, HW, 3*HW, HW, 3*HW, HW, 128, 128, 128, 128,
      f_D, x, 3*HW, HW, nullptr, nullptr);
  // Pa = W @ G + b
  gemm128<2><<<dim3(8, 128), 256, 0, stream>>>(f_W, Pb, Pa, 1,
      0, 0, 3*HW, 0, 128*HW, 0, 3, 16384, 16384, 3,
      nullptr, nullptr, 0, 0, f_b, nullptr);

  // ---- 4 residual layers: cin = 128 ----
  for (int i = 0; i < 4; ++i) {
    const float* ld = l_log_dt + (long)i * 2 * 128;
    const float* la = l_logA + (long)i * 8192;
    const float* ai = l_Aim  + (long)i * 8192;
    const float* pr = l_Pre  + (long)i * 8192;
    const float* pi = l_Pim  + (long)i * 8192;
    const float* br = l_Bre  + (long)i * 8192;
    const float* bi = l_Bim  + (long)i * 8192;
    const float* cr = l_Cre  + (long)i * 16384;
    const float* ci = l_Cim  + (long)i * 16384;
    const float* dd = l_D    + (long)i * 128;
    const float* wm = l_W    + (long)i * 16384;
    const float* bb = l_b    + (long)i * 128;

    kern_build<<<dim3(128, 2), 256, 0, stream>>>(ld, la, ai, pr, pi, br, bi, cr, ci,
                                                 128, khh, khw);
    mat_build<<<128, 256, 0, stream>>>(khh, khw, Mh, MwT);
    // T(Pb) = X(Pa) @ MwT
    gemm128<0><<<dim3(1024, 1), 256, 0, stream>>>(Pa, MwT, Pb, 128,
        128*HW, HW, 0, HW, 128*HW, HW, 128, 128, 128, 128,
        nullptr, nullptr, 0, 0, nullptr, nullptr);
    // Pb = gelu(Mh @ Pb + D*Pa)   (in-place per-tile)
    gemm128<1><<<dim3(1024, 1), 256, 0, stream>>>(Mh, Pb, Pb, 128,
        0, HW, 128*HW, HW, 128*HW, HW, 128, 128, 128, 128,
        dd, Pa, 128*HW, HW, nullptr, nullptr);
    // Pa = W @ G + b + Pa  (residual; element-exact read-then-write)
    gemm128<2><<<dim3(8, 128), 256, 0, stream>>>(wm, Pb, Pa, 1,
        0, 0, 128*HW, 0, 128*HW, 0, 128, 16384, 16384, 128,
        nullptr, nullptr, 0, 0, bb, Pa);
    // postnorm
    bn2d<<<128, 256, 0, stream>>>(Pa, bn_g + (long)i * 128, bn_b + (long)i * 128);
  }

  pool_mean<<<1024, 256, 0, stream>>>(Pa, (float*)d_out);
}